// InteractionGNN_12326556139999
// MI455X (gfx1250) — compile-verified
//
#include <hip/hip_runtime.h>
#include <hip/hip_bf16.h>
#include <math.h>

typedef float v2f __attribute__((ext_vector_type(2)));
typedef float v8f __attribute__((ext_vector_type(8)));

#define HID 128

// ---- LDS helpers: generic -> addrspace(3) offset for async-to-LDS ops ----
typedef __attribute__((address_space(3))) const void lds_cv;
__device__ __forceinline__ unsigned lds_off(const void* p) {
  return (unsigned)(uintptr_t)(lds_cv*)p;
}

// One instruction: each active lane copies 16 bytes from global memory at base plus its voff to the LDS address in its lds_addr.
// Tracked by ASYNCcnt (CDNA5 async global->LDS path).
__device__ __forceinline__ void async_copy_b128(unsigned lds_addr, unsigned voff,
                                                const void* base) {
  asm volatile("global_load_async_to_lds_b128 %0, %1, %2"
               :: "v"(lds_addr), "v"(voff), "s"(base)
               : "memory");
}
__device__ __forceinline__ void wait_async0() {
  asm volatile("s_wait_asynccnt 0x0" ::: "memory");
}

// ---------------- degree / normalization ----------------

__global__ __launch_bounds__(256) void k_deg_init(float* __restrict__ deg, int n) {
  int i = blockIdx.x * 256 + threadIdx.x;
  if (i < n) deg[i] = 1.0f;  // self-loop contributes 1
}

__global__ __launch_bounds__(256) void k_deg_accum(const int* __restrict__ col,
                                                   float* __restrict__ deg, int e) {
  int i = blockIdx.x * 256 + threadIdx.x;
  if (i < e)
    __hip_atomic_fetch_add(&deg[col[i]], 1.0f, __ATOMIC_RELAXED, __HIP_MEMORY_SCOPE_AGENT);
}

__global__ __launch_bounds__(256) void k_dinv(float* __restrict__ deg, int n) {
  int i = blockIdx.x * 256 + threadIdx.x;
  if (i < n) deg[i] = rsqrtf(deg[i]);  // deg >= 1 always
}

// ---------------- 16xHID row-tile GEMM: Y[16,128] = A[16,128] @ W[128,128] ----------------
// 8 waves per block; wave w computes the 16x16 tile at columns [16w, 16w+16).
// A tile staged in LDS via async b128 copies (stride 132 floats -> no bank conflicts,
// rows stay 16B aligned: 132*4 = 528 = 33*16).

__global__ __launch_bounds__(256) void k_gemm128(const float* __restrict__ A,
                                                 const float* __restrict__ W,
                                                 float* __restrict__ Y, int n) {
  __shared__ float As[16 * 132];
  const int tid  = threadIdx.x;
  const int wave = tid >> 5;
  const int lane = tid & 31;
  const int r0   = blockIdx.x * 16;

  const float* Ab = A + (size_t)r0 * HID;
  __builtin_prefetch(Ab + 16 * HID, 0, 0);  // speculative prefetch of next tile

  // stage A tile: 512 float4 chunks, async global->LDS
  for (int idx = tid; idx < 16 * 32; idx += 256) {
    int r = idx >> 5, q = idx & 31;           // row, float4-within-row
    if (r0 + r < n) {
      unsigned voff = (unsigned)(r * 512 + q * 16);
      async_copy_b128(lds_off(&As[r * 132 + q * 4]), voff, Ab);
    } else {
      float4 z = {0.f, 0.f, 0.f, 0.f};
      *(float4*)&As[r * 132 + q * 4] = z;
    }
  }
  wait_async0();
  __syncthreads();

  const int m     = lane & 15;          // A row within tile
  const int ncol  = wave * 16 + m;      // output column
  const int khalf = (lane >> 4) * 2;    // lanes 16-31 hold K+2,K+3

  v8f acc = {};
  for (int k0 = 0; k0 < HID; k0 += 4) {
    const int kk = k0 + khalf;
    v2f a; a.x = As[m * 132 + kk];       a.y = As[m * 132 + kk + 1];
    v2f b; b.x = W[kk * HID + ncol];     b.y = W[(kk + 1) * HID + ncol];
    acc = __builtin_amdgcn_wmma_f32_16x16x4_f32(false, a, false, b,
                                                (short)0, acc, false, false);
  }

  const int mbase = (lane >> 4) * 8;    // D: vgpr i -> M=i (lo lanes) / i+8 (hi lanes)
  float* Yb = Y + (size_t)r0 * HID;
#pragma unroll
  for (int i = 0; i < 8; ++i) {
    int mm = mbase + i;
    if (r0 + mm < n) Yb[(size_t)mm * HID + ncol] = acc[i];
  }
}

// ---------------- self-loop init: h = t * dinv^2 ----------------

__global__ __launch_bounds__(256) void k_selfinit(const float* __restrict__ t,
                                                  const float* __restrict__ dinv,
                                                  float* __restrict__ h, int n) {
  int idx = blockIdx.x * 256 + threadIdx.x;
  if (idx < n * HID) {
    float d = dinv[idx >> 7];
    h[idx] = t[idx] * d * d;
  }
}

// ---------------- edge scatter: h[col] += t[row] * dinv[row]*dinv[col] ----------------
// one wave32 per edge, 4 features per lane, native f32 global atomics

__global__ __launch_bounds__(256) void k_scatter(const int* __restrict__ row,
                                                 const int* __restrict__ col,
                                                 const float* __restrict__ t,
                                                 const float* __restrict__ dinv,
                                                 float* __restrict__ h, int e) {
  int gid  = blockIdx.x * 256 + threadIdx.x;
  int eid  = gid >> 5;
  int lane = gid & 31;
  if (eid < e) {
    int r = row[eid], c = col[eid];
    float w = dinv[r] * dinv[c];
    const float* ts = t + (size_t)r * HID;
    float*       hd = h + (size_t)c * HID;
#pragma unroll
    for (int j = 0; j < 4; ++j) {
      int f = lane + 32 * j;
      __hip_atomic_fetch_add(&hd[f], ts[f] * w, __ATOMIC_RELAXED, __HIP_MEMORY_SCOPE_AGENT);
    }
  }
}

__global__ __launch_bounds__(256) void k_bias_relu(float* __restrict__ h,
                                                   const float* __restrict__ b, int n) {
  int idx = blockIdx.x * 256 + threadIdx.x;
  if (idx < n * HID) {
    float v = h[idx] + b[idx & 127];
    h[idx] = v > 0.0f ? v : 0.0f;
  }
}

// ---------------- fused edge MLP ----------------
// 16 edges/block. ef[16,256] = concat(h[row], h[col]) gathered straight into LDS with
// per-lane async b128 copies (stride 260 floats keeps 16B alignment: 1040 = 65*16).
// z = relu(ef @ We1 + be1) kept in registers (16x16 tiles over 8 waves),
// p = sigmoid(z @ We2 + be2) via LDS f32-atomic reduction over the 128 hidden cols.

__global__ __launch_bounds__(256) void k_edge_mlp(const int* __restrict__ row,
                                                  const int* __restrict__ col,
                                                  const float* __restrict__ h,
                                                  const float* __restrict__ We1,
                                                  const float* __restrict__ be1,
                                                  const float* __restrict__ We2,
                                                  const float* __restrict__ be2,
                                                  float* __restrict__ out, int E) {
  __shared__ float ef[16 * 260];
  __shared__ float part[16];
  const int tid = threadIdx.x;
  const int e0  = blockIdx.x * 16;

  if (tid < 16) part[tid] = 0.0f;

  // gather: 16 edges x 64 float4 (32 from h[row], 32 from h[col]) -> async to LDS
  for (int idx = tid; idx < 16 * 64; idx += 256) {
    int le = idx >> 6;          // edge within tile
    int q  = idx & 63;          // float4 index within 256-float concat row
    int e  = e0 + le;
    if (e >= E) e = E - 1;
    int node = (q < 32) ? row[e] : col[e];
    unsigned voff = (unsigned)node * 512u + (unsigned)(q & 31) * 16u;
    async_copy_b128(lds_off(&ef[le * 260 + q * 4]), voff, h);
  }
  wait_async0();
  __syncthreads();

  const int wave  = tid >> 5;
  const int lane  = tid & 31;
  const int m     = lane & 15;          // edge within the 16-edge tile
  const int ncol  = wave * 16 + m;      // hidden column
  const int khalf = (lane >> 4) * 2;

  v8f acc = {};
  for (int k0 = 0; k0 < 2 * HID; k0 += 4) {
    const int kk = k0 + khalf;
    v2f a; a.x = ef[m * 260 + kk];        a.y = ef[m * 260 + kk + 1];
    v2f b; b.x = We1[kk * HID + ncol];    b.y = We1[(kk + 1) * HID + ncol];
    acc = __builtin_amdgcn_wmma_f32_16x16x4_f32(false, a, false, b,
                                                (short)0, acc, false, false);
  }

  const float bias  = be1[ncol];
  const float w2    = We2[ncol];
  const int   mbase = (lane >> 4) * 8;
#pragma unroll
  for (int i = 0; i < 8; ++i) {
    float z = acc[i] + bias;
    z = z > 0.0f ? z : 0.0f;               // relu
    __hip_atomic_fetch_add(&part[mbase + i], z * w2,
                           __ATOMIC_RELAXED, __HIP_MEMORY_SCOPE_WORKGROUP);
  }
  __syncthreads();

  if (tid < 16 && (e0 + tid) < E) {
    float s = part[tid] + be2[0];
    out[e0 + tid] = 1.0f / (1.0f + __expf(-s));
  }
}

// ---------------- launch ----------------

extern "C" void kernel_launch(void* const* d_in, const int* in_sizes, int n_in,
                              void* d_out, int out_size, void* d_ws, size_t ws_size,
                              hipStream_t stream) {
  const float* x   = (const float*)d_in[0];
  const int*   ei  = (const int*)  d_in[1];
  const float* W1  = (const float*)d_in[2];
  const float* b1  = (const float*)d_in[3];
  const float* W2  = (const float*)d_in[4];
  const float* b2  = (const float*)d_in[5];
  const float* We1 = (const float*)d_in[6];
  const float* be1 = (const float*)d_in[7];
  const float* We2 = (const float*)d_in[8];
  const float* be2 = (const float*)d_in[9];
  float* out = (float*)d_out;

  const int N = in_sizes[0] / HID;
  const int E = in_sizes[1] / 2;
  const int* row = ei;
  const int* col = ei + E;

  float* ws   = (float*)d_ws;
  float* dinv = ws;
  float* tA   = ws + (size_t)((N + 3) & ~3);   // GEMM output (pre-scatter)
  float* hB   = tA + (size_t)N * HID;          // post-scatter hidden state

  int nbN  = (N + 255) / 256;
  int nbE  = (E + 255) / 256;
  int nbNH = (N * HID + 255) / 256;
  int nbEW = ((E * 32) + 255) / 256;           // one wave per edge
  int rowTiles = (N + 15) / 16;
  int edgeTiles = (E + 15) / 16;

  // normalization coefficients
  k_deg_init <<<nbN, 256, 0, stream>>>(dinv, N);
  k_deg_accum<<<nbE, 256, 0, stream>>>(col, dinv, E);
  k_dinv     <<<nbN, 256, 0, stream>>>(dinv, N);

  // GCN layer 1
  k_gemm128  <<<rowTiles, 256, 0, stream>>>(x, W1, tA, N);
  k_selfinit <<<nbNH, 256, 0, stream>>>(tA, dinv, hB, N);
  k_scatter  <<<nbEW, 256, 0, stream>>>(row, col, tA, dinv, hB, E);
  k_bias_relu<<<nbNH, 256, 0, stream>>>(hB, b1, N);

  // GCN layer 2 (tA reused; hB rewritten after being consumed by the GEMM)
  k_gemm128  <<<rowTiles, 256, 0, stream>>>(hB, W2, tA, N);
  k_selfinit <<<nbNH, 256, 0, stream>>>(tA, dinv, hB, N);
  k_scatter  <<<nbEW, 256, 0, stream>>>(row, col, tA, dinv, hB, E);
  k_bias_relu<<<nbNH, 256, 0, stream>>>(hB, b2, N);

  // fused edge MLP -> sigmoid probabilities
  k_edge_mlp <<<edgeTiles, 256, 0, stream>>>(row, col, hB, We1, be1, We2, be2, out, E);
}